// RumorGNN_30305289241271
// MI455X (gfx1250) — compile-verified
//
#include <hip/hip_runtime.h>
#include <math.h>
#include <stdint.h>

#define IN_DIM    128
#define HID       256
#define HEADS     4
#define CPH       64
#define NCLS      2
#define NEG_SLOPE 0.2f
#define BN_EPS    1e-5f

#define GEMM_BM   128      // rows per block  (8 waves x 16)
#define GEMM_BN   64       // cols per block
#define GEMM_BSTR 72       // LDS row stride (floats): 64 + 8 pad -> conflict-free
#define GEMM_MAXK 256      // max K staged in LDS (256*72*4B = 72 KB)

typedef float v2f __attribute__((ext_vector_type(2)));
typedef float v8f __attribute__((ext_vector_type(8)));

// ---------------------------------------------------------------------------
// WMMA f32 GEMM:  C[M x Nc] = A[M x K] @ B[K x Nc], all row-major fp32.
//
// Block = 256 threads = 8 waves computing a 128x64 tile:
//   - B[0:K, n0:n0+64] staged into LDS (row stride 72 floats for bank-conflict
//     freedom) via CDNA5 async global->LDS 16B copies (ASYNCcnt).  The LDS
//     destination pointer is passed into the asm so the LDS global escapes and
//     the compiler cannot fold the subsequent Bs loads away.
//   - Each wave: 16x64 strip, 4 f32 accumulators, K stepped by 4 with
//     V_WMMA_F32_16X16X4_F32 (exact fp32, matches the fp32 reference).
//   - A fragment for the next K-step is loaded unconditionally (index select,
//     no branch) so the single global_load_b64 overlaps the 4 WMMAs.
//
// VGPR layouts per CDNA5 ISA 7.12.2:
//   A 16x4 : lanes 0-15 -> K={0,1}, lanes 16-31 -> K={2,3}, row = lane%16
//   B 4x16 : .x -> row koff, .y -> row koff+1, col = lane%16
//   C 16x16: vgpr i -> row i (lanes 0-15) / row i+8 (lanes 16-31), col lane%16
// Requires Nc % 64 == 0, K % 4 == 0, K <= GEMM_MAXK (true here: 128/256).
// ---------------------------------------------------------------------------
__global__ __launch_bounds__(256)
void gemm_wmma_f32(const float* __restrict__ A, const float* __restrict__ B,
                   float* __restrict__ C, int M, int K, int Nc) {
  __shared__ float Bs[GEMM_MAXK * GEMM_BSTR];

  const int tid  = threadIdx.x;
  const int lane = tid & 31;
  const int wave = tid >> 5;
  const int m0   = blockIdx.x * GEMM_BM + wave * 16;
  const int n0   = blockIdx.y * GEMM_BN;

  // ---- Stage B tile into LDS with async 16B copies (tracked by ASYNCcnt).
  // Bs[k*72 + j .. +3] = B[k*Nc + n0 + j .. +3];  16 chunks per K-row.
  const int nchunks = K << 4;                  // K rows * 16 float4 chunks
  for (int idx = tid; idx < nchunks; idx += 256) {
    const int k = idx >> 4;
    const int j = (idx & 15) << 2;
    float* dst = &Bs[(size_t)k * GEMM_BSTR + j];
    const unsigned lds_byte = (unsigned)(uintptr_t)dst;   // low 32b = LDS offset
    const unsigned long long gaddr =
        (unsigned long long)(const void*)(B + (size_t)k * Nc + n0 + j);
    asm volatile("global_load_async_to_lds_b128 %0, %1, off"
                 :: "v"(lds_byte), "v"(gaddr), "v"(dst) : "memory");
  }
  asm volatile("s_wait_asynccnt 0x0" ::: "memory");
  __syncthreads();

  // ---- Per-wave 16x64 WMMA strip.
  const int r16  = lane & 15;
  const int koff = (lane >> 4) << 1;           // 0 for lanes 0-15, 2 for 16-31
  int arow = m0 + r16;                          // clamp reads; stores guarded
  if (arow >= M) arow = M - 1;
  const float* __restrict__ ap = A + (size_t)arow * K;

  v8f acc0 = {}, acc1 = {}, acc2 = {}, acc3 = {};
  v2f a_cur;
  a_cur.x = ap[koff];
  a_cur.y = ap[koff + 1];

  for (int k = 0; k < K; k += 4) {
    // Branch-free prefetch of the next A fragment (overlaps the 4 WMMAs).
    const int kn = (k + 4 < K) ? (k + 4) : 0;   // uniform index select
    v2f a_nxt;
    a_nxt.x = ap[kn + koff];
    a_nxt.y = ap[kn + koff + 1];

    const float* bs0 = &Bs[(size_t)(k + koff) * GEMM_BSTR + r16];
    const float* bs1 = bs0 + GEMM_BSTR;
    v2f b;
    b.x = bs0[0];  b.y = bs1[0];
    acc0 = __builtin_amdgcn_wmma_f32_16x16x4_f32(false, a_cur, false, b, (short)0, acc0, false, false);
    b.x = bs0[16]; b.y = bs1[16];
    acc1 = __builtin_amdgcn_wmma_f32_16x16x4_f32(false, a_cur, false, b, (short)0, acc1, false, false);
    b.x = bs0[32]; b.y = bs1[32];
    acc2 = __builtin_amdgcn_wmma_f32_16x16x4_f32(false, a_cur, false, b, (short)0, acc2, false, false);
    b.x = bs0[48]; b.y = bs1[48];
    acc3 = __builtin_amdgcn_wmma_f32_16x16x4_f32(false, a_cur, false, b, (short)0, acc3, false, false);
    a_cur = a_nxt;
  }

  const int crow0 = m0 + ((lane >> 4) << 3);
  const int ccol  = n0 + r16;
#pragma unroll
  for (int i = 0; i < 8; ++i) {
    const int r = crow0 + i;
    if (r < M) {
      float* cp = C + (size_t)r * Nc + ccol;
      cp[0]  = acc0[i];
      cp[16] = acc1[i];
      cp[32] = acc2[i];
      cp[48] = acc3[i];
    }
  }
}

// Tiny-N GEMM for the classifier head (Nc == 2).
__global__ void gemm_n2(const float* __restrict__ A, const float* __restrict__ W,
                        float* __restrict__ C, int M, int K) {
  const int n = blockIdx.x * blockDim.x + threadIdx.x;
  if (n >= M) return;
  const float* ap = A + (size_t)n * K;
  float a0 = 0.f, a1 = 0.f;
  for (int k = 0; k < K; ++k) {
    const float av = ap[k];
    a0 += av * W[2 * k];
    a1 += av * W[2 * k + 1];
  }
  C[2 * n]     = a0;
  C[2 * n + 1] = a1;
}

// alpha_src[n,h] = sum_c h[n,h,c]*a_src[h,c]  (and same for a_dst)
__global__ void node_alpha(const float* __restrict__ h,
                           const float* __restrict__ a_s, const float* __restrict__ a_d,
                           float* __restrict__ as_out, float* __restrict__ ad_out,
                           int N, int H, int C) {
  const int tid = blockIdx.x * blockDim.x + threadIdx.x;
  if (tid >= N * H) return;
  const int n = tid / H, hh = tid % H;
  const float* hp  = h + (size_t)n * H * C + (size_t)hh * C;
  const float* asp = a_s + (size_t)hh * C;
  const float* adp = a_d + (size_t)hh * C;
  float s0 = 0.f, s1 = 0.f;
  for (int c = 0; c < C; ++c) {
    const float hv = hp[c];
    s0 += hv * asp[c];
    s1 += hv * adp[c];
  }
  as_out[tid] = s0;
  ad_out[tid] = s1;
}

__global__ void fill_f32(float* __restrict__ p, float v, int n) {
  const int i = blockIdx.x * blockDim.x + threadIdx.x;
  if (i < n) p[i] = v;
}

// Float atomic max via sign-split integer atomics (valid with -inf init).
__device__ __forceinline__ void atomicMaxF(float* addr, float val) {
  if (val >= 0.f) atomicMax((int*)addr, __float_as_int(val));
  else            atomicMin((unsigned int*)addr, __float_as_uint(val));
}

__device__ __forceinline__ void edge_endpoints(const int* __restrict__ ei,
                                               int E, int e, int& s, int& d) {
  if (e < E) { s = ei[e]; d = ei[(size_t)E + e]; }
  else       { s = d = e - E; }                       // self-loops appended
}

// Pass 1: e = leakyrelu(as[src]+ad[dst]); segment max into nmax[dst,h]
__global__ void edge_score_max(const int* __restrict__ ei, int E, int N, int H,
                               const float* __restrict__ as, const float* __restrict__ ad,
                               float* __restrict__ nmax) {
  const long tid = (long)blockIdx.x * blockDim.x + threadIdx.x;
  const long total = (long)(E + N) * H;
  if (tid >= total) return;
  const int e = (int)(tid / H), hh = (int)(tid % H);
  int s, d; edge_endpoints(ei, E, e, s, d);
  float v = as[(size_t)s * H + hh] + ad[(size_t)d * H + hh];
  v = v > 0.f ? v : NEG_SLOPE * v;
  atomicMaxF(&nmax[(size_t)d * H + hh], v);
}

// Pass 2: ex = exp(e - max); store ex, segment-sum into den[dst,h]
__global__ void edge_exp_sum(const int* __restrict__ ei, int E, int N, int H,
                             const float* __restrict__ as, const float* __restrict__ ad,
                             const float* __restrict__ nmax,
                             float* __restrict__ exb, float* __restrict__ den) {
  const long tid = (long)blockIdx.x * blockDim.x + threadIdx.x;
  const long total = (long)(E + N) * H;
  if (tid >= total) return;
  const int e = (int)(tid / H), hh = (int)(tid % H);
  int s, d; edge_endpoints(ei, E, e, s, d);
  float v = as[(size_t)s * H + hh] + ad[(size_t)d * H + hh];
  v = v > 0.f ? v : NEG_SLOPE * v;
  const float ex = __expf(v - nmax[(size_t)d * H + hh]);
  exb[(size_t)e * H + hh] = ex;
  atomicAdd(&den[(size_t)d * H + hh], ex);
}

__global__ void invert_den(float* __restrict__ den, int n) {
  const int i = blockIdx.x * blockDim.x + threadIdx.x;
  if (i < n) den[i] = 1.f / (den[i] + 1e-16f);
}

// Pass 3: out[dst] += h[src] * alpha ;  CHUNK channels per thread.
template <int CHUNK>
__global__ void edge_aggregate(const int* __restrict__ ei, int E, int N, int H, int C,
                               const float* __restrict__ hfeat,
                               const float* __restrict__ exb,
                               const float* __restrict__ deninv,
                               float* __restrict__ out) {
  const int HC = H * C;
  const int chunks = HC / CHUNK;
  const long tid = (long)blockIdx.x * blockDim.x + threadIdx.x;
  const long total = (long)(E + N) * chunks;
  if (tid >= total) return;
  const int e  = (int)(tid / chunks);
  const int ch = (int)(tid % chunks) * CHUNK;
  int s, d; edge_endpoints(ei, E, e, s, d);
  const int head = ch / C;
  const float alpha = exb[(size_t)e * H + head] * deninv[(size_t)d * H + head];
  const float* hp = hfeat + (size_t)s * HC + ch;
  float* op = out + (size_t)d * HC + ch;
#pragma unroll
  for (int i = 0; i < CHUNK; ++i) atomicAdd(op + i, hp[i] * alpha);
}

// y = relu( ((agg + bias) - mu) * g * rsqrt(var+eps) + beta )
__global__ void bias_bn_relu(const float* __restrict__ agg, const float* __restrict__ b,
                             const float* __restrict__ g, const float* __restrict__ be,
                             const float* __restrict__ mu, const float* __restrict__ var,
                             float* __restrict__ out, long total, int HC) {
  const long tid = (long)blockIdx.x * blockDim.x + threadIdx.x;
  if (tid >= total) return;
  const int j = (int)(tid % HC);
  const float y = (agg[tid] + b[j] - mu[j]) * (g[j] * rsqrtf(var[j] + BN_EPS)) + be[j];
  out[tid] = fmaxf(y, 0.f);
}

// out = log_softmax(agg + b2) over 2 classes
__global__ void final_logsoftmax(const float* __restrict__ agg, const float* __restrict__ b2,
                                 float* __restrict__ out, int N) {
  const int n = blockIdx.x * blockDim.x + threadIdx.x;
  if (n >= N) return;
  const float x0 = agg[2 * n]     + b2[0];
  const float x1 = agg[2 * n + 1] + b2[1];
  const float mx  = fmaxf(x0, x1);
  const float lse = mx + logf(expf(x0 - mx) + expf(x1 - mx));
  out[2 * n]     = x0 - lse;
  out[2 * n + 1] = x1 - lse;
}

// ---------------------------------------------------------------------------
// Host-side: one GAT layer (GEMM + attention softmax + scatter aggregation)
// ---------------------------------------------------------------------------
static void run_gat_layer(const float* xin, const float* W,
                          const float* a_s, const float* a_d,
                          const int* ei, int E, int N, int H, int C, int K,
                          float* hbuf, float* aggbuf,
                          float* asrc, float* adst, float* nmax, float* den, float* exb,
                          bool useWmma, hipStream_t stream) {
  const int HC = H * C;
  if (useWmma) {
    dim3 grid((N + GEMM_BM - 1) / GEMM_BM, HC / GEMM_BN);
    gemm_wmma_f32<<<grid, 256, 0, stream>>>(xin, W, hbuf, N, K, HC);
  } else {
    gemm_n2<<<(N + 255) / 256, 256, 0, stream>>>(xin, W, hbuf, N, K);
  }

  const int nh = N * H;
  node_alpha<<<(nh + 255) / 256, 256, 0, stream>>>(hbuf, a_s, a_d, asrc, adst, N, H, C);
  fill_f32<<<(nh + 255) / 256, 256, 0, stream>>>(nmax, -INFINITY, nh);
  hipMemsetAsync(den, 0, (size_t)nh * sizeof(float), stream);

  const long te = (long)(E + N) * H;
  edge_score_max<<<(int)((te + 255) / 256), 256, 0, stream>>>(ei, E, N, H, asrc, adst, nmax);
  edge_exp_sum<<<(int)((te + 255) / 256), 256, 0, stream>>>(ei, E, N, H, asrc, adst, nmax, exb, den);
  invert_den<<<(nh + 255) / 256, 256, 0, stream>>>(den, nh);

  hipMemsetAsync(aggbuf, 0, (size_t)N * HC * sizeof(float), stream);
  if ((HC & 3) == 0) {
    const long ta = (long)(E + N) * (HC / 4);
    edge_aggregate<4><<<(int)((ta + 255) / 256), 256, 0, stream>>>(ei, E, N, H, C, hbuf, exb, den, aggbuf);
  } else {
    const long ta = (long)(E + N) * (HC / 2);
    edge_aggregate<2><<<(int)((ta + 255) / 256), 256, 0, stream>>>(ei, E, N, H, C, hbuf, exb, den, aggbuf);
  }
}

extern "C" void kernel_launch(void* const* d_in, const int* in_sizes, int n_in,
                              void* d_out, int out_size, void* d_ws, size_t ws_size,
                              hipStream_t stream) {
  const float* x     = (const float*)d_in[0];
  const int*   ei    = (const int*)  d_in[1];
  const float* W0    = (const float*)d_in[2];
  const float* as0   = (const float*)d_in[3];
  const float* ad0   = (const float*)d_in[4];
  const float* b0    = (const float*)d_in[5];
  const float* g0    = (const float*)d_in[6];
  const float* be0   = (const float*)d_in[7];
  const float* m0    = (const float*)d_in[8];
  const float* v0    = (const float*)d_in[9];
  const float* W1    = (const float*)d_in[10];
  const float* as1   = (const float*)d_in[11];
  const float* ad1   = (const float*)d_in[12];
  const float* b1    = (const float*)d_in[13];
  const float* g1    = (const float*)d_in[14];
  const float* be1   = (const float*)d_in[15];
  const float* m1    = (const float*)d_in[16];
  const float* v1    = (const float*)d_in[17];
  const float* W2    = (const float*)d_in[18];
  const float* as2   = (const float*)d_in[19];
  const float* ad2   = (const float*)d_in[20];
  const float* b2    = (const float*)d_in[21];

  const int N = in_sizes[0] / IN_DIM;      // 50000
  const int E = in_sizes[1] / 2;           // 800000 (before self-loops)

  // Workspace layout (floats)
  float* ws = (float*)d_ws;
  size_t off = 0;
  float* bufA = ws + off; off += (size_t)N * HID;         // features / BN out
  float* bufB = ws + off; off += (size_t)N * HID;         // GEMM out / agg
  float* asrc = ws + off; off += (size_t)N * HEADS;
  float* adst = ws + off; off += (size_t)N * HEADS;
  float* nmax = ws + off; off += (size_t)N * HEADS;
  float* den  = ws + off; off += (size_t)N * HEADS;
  float* exb  = ws + off; off += (size_t)(E + N) * HEADS;
  float* h2   = ws + off; off += (size_t)N * NCLS;
  float* agg2 = ws + off; off += (size_t)N * NCLS;
  (void)ws_size; (void)n_in;

  const long tot = (long)N * HID;
  const int  ebn = (int)((tot + 255) / 256);

  // ---- Layer 0: h0 -> bufB (WMMA), agg0 -> bufA, BN(bufA) -> bufB
  run_gat_layer(x,   W0, as0, ad0, ei, E, N, HEADS, CPH, IN_DIM,
                /*h*/bufB, /*agg*/bufA, asrc, adst, nmax, den, exb, true, stream);
  bias_bn_relu<<<ebn, 256, 0, stream>>>(bufA, b0, g0, be0, m0, v0, bufB, tot, HID);

  // ---- Layer 1: h1 = bufB@W1 -> bufA, agg1 -> bufB (free after GEMM reads,
  //      stream-serialized), BN(bufB) -> bufA
  run_gat_layer(bufB, W1, as1, ad1, ei, E, N, HEADS, CPH, HID,
                /*h*/bufA, /*agg*/bufB, asrc, adst, nmax, den, exb, true, stream);
  bias_bn_relu<<<ebn, 256, 0, stream>>>(bufB, b1, g1, be1, m1, v1, bufA, tot, HID);

  // ---- Layer 2: heads=1, channels=NCLS
  run_gat_layer(bufA, W2, as2, ad2, ei, E, N, 1, NCLS, HID,
                /*h*/h2, /*agg*/agg2, asrc, adst, nmax, den, exb, false, stream);

  final_logsoftmax<<<(N + 255) / 256, 256, 0, stream>>>(agg2, b2, (float*)d_out, N);
}